// ForwardKinematicsModule_66984309949100
// MI455X (gfx1250) — compile-verified
//
#include <hip/hip_runtime.h>
#include <math.h>

#define TPB 256
#define BPT 2                    // bodies per thread (packed f32 pairs)
#define TILE (TPB * BPT)         // 512 bodies per block

typedef int   v4i __attribute__((ext_vector_type(4)));
typedef float f2  __attribute__((ext_vector_type(2)));
#define AS1 __attribute__((address_space(1)))
#define AS3 __attribute__((address_space(3)))

#if __has_builtin(__builtin_amdgcn_global_load_async_to_lds_b128) && \
    __has_builtin(__builtin_amdgcn_global_store_async_from_lds_b128) && \
    __has_builtin(__builtin_amdgcn_s_wait_asynccnt)
#define USE_ASYNC 1
#else
#define USE_ASYNC 0
#endif

__device__ __forceinline__ void quat_from_R(float r00, float r01, float r02,
                                            float r10, float r11, float r12,
                                            float r20, float r21, float r22,
                                            float& qw, float& qx, float& qy, float& qz)
{
    const float trace = r00 + r11 + r22;
    const bool  m  = trace > 0.0f;
    const float s  = 2.0f * __builtin_sqrtf(fmaxf(trace, 0.0f) + 1.0f);
    const float rs = __builtin_amdgcn_rcpf(s);
    qw = m ? 0.25f * s : 0.0f;
    qx = m ? (r21 - r12) * rs : 0.0f;
    qy = m ? (r02 - r20) * rs : 0.0f;
    qz = m ? (r10 - r01) * rs : 0.0f;
}

// Forward kinematics, 6-DOF DH chain. Two bodies per lane -> v_pk_fma_f32 path.
// ~218 MB streamed -> ~9.4 us HBM floor @ 23.3 TB/s; packed math keeps VALU under it.
__global__ __launch_bounds__(TPB) void fk_kernel(
    const float* __restrict__ ja,   // [B,6]
    const float* __restrict__ a6,   // [6]
    const float* __restrict__ d6,   // [6]
    const float* __restrict__ al6,  // [6]
    float* __restrict__ out,        // [B,7] pos(3)+quat wxyz(4)
    int B)
{
    __shared__ __align__(16) float lds[TILE * 7];  // 14 KB, in-stage then out-stage

    const int tid = threadIdx.x;
    const int tileBase = blockIdx.x * TILE;
    const bool fullTile = (tileBase + TILE) <= B;

    const long long inBase = (long long)tileBase * 6;
    const float* __restrict__ gin = ja + inBase;

    // ---- stage inputs: 3072 floats = 768 float4 = 3 b128 per thread, exact ----
    if (fullTile) {
#if USE_ASYNC
#pragma unroll
        for (int i = 0; i < 3; ++i) {
            const int f4 = i * TPB + tid;
            __builtin_amdgcn_global_load_async_to_lds_b128(
                (AS1 v4i*)(gin + 4 * f4), (AS3 v4i*)(lds + 4 * f4), 0, 0);
        }
        __builtin_amdgcn_s_wait_asynccnt(0);
#else
#pragma unroll
        for (int i = 0; i < 3; ++i) {
            const int f4 = i * TPB + tid;
            ((float4*)lds)[f4] = ((const float4*)gin)[f4];
        }
#endif
    } else {
        const int nIn = (B - tileBase) * 6;
#pragma unroll
        for (int i = 0; i < 12; ++i) {
            const int f = i * TPB + tid;
            if (f < nIn) lds[f] = gin[f];
        }
    }

    // Warm shared 192 MB L2 with next tile (gfx1250 global_prefetch_b8).
    {
        const long long nf = inBase + (long long)TILE * 6 + (long long)tid * 12;
        if (nf < (long long)B * 6) __builtin_prefetch(&ja[nf], 0, 1);
    }

    __syncthreads();

    // Bodies 2*tid and 2*tid+1: 12 contiguous floats -> 3x ds_load_b128.
    float w[12];
    {
        const float4* lw = (const float4*)(lds + tid * 12);
        const float4 A = lw[0], C = lw[1], E = lw[2];
        w[0]=A.x; w[1]=A.y; w[2]=A.z; w[3]=A.w;
        w[4]=C.x; w[5]=C.y; w[6]=C.z; w[7]=C.w;
        w[8]=E.x; w[9]=E.y; w[10]=E.z; w[11]=E.w;
    }
    f2 th[6];
#pragma unroll
    for (int i = 0; i < 6; ++i) { th[i].x = w[i]; th[i].y = w[6 + i]; }

    __syncthreads();  // input reads done before LDS reuse

    // Uniform DH parameters -> s_load_b32.
    float av[6], dv[6], alv[6];
#pragma unroll
    for (int i = 0; i < 6; ++i) { av[i] = a6[i]; dv[i] = d6[i]; alv[i] = al6[i]; }

    // Packed transform: rows 0..2 of 4x4, two bodies in .x/.y lanes.
    f2 R[3][4];
#pragma unroll
    for (int r = 0; r < 3; ++r)
#pragma unroll
        for (int c = 0; c < 4; ++c) R[r][c] = (r == c) ? 1.0f : 0.0f;

#pragma unroll
    for (int i = 0; i < 6; ++i) {
        float s0, c0, s1, c1;
        __sincosf(th[i].x, &s0, &c0);
        __sincosf(th[i].y, &s1, &c1);
        f2 st; st.x = s0; st.y = s1;
        f2 ct; ct.x = c0; ct.y = c1;
        float saf, caf;
        __sincosf(alv[i], &saf, &caf);
        const f2 sa = saf, ca = caf, ai = av[i], di = dv[i];
#pragma unroll
        for (int r = 0; r < 3; ++r) {
            const f2 p0 = R[r][0], p1 = R[r][1], p2 = R[r][2], p3 = R[r][3];
            const f2 u = __builtin_elementwise_fma(p0, ct, p1 * st);
            const f2 v = __builtin_elementwise_fma(p1, ct, -(p0 * st));
            R[r][0] = u;
            R[r][1] = __builtin_elementwise_fma(ca,  v, p2 * sa);
            R[r][2] = __builtin_elementwise_fma(-sa, v, p2 * ca);
            R[r][3] = __builtin_elementwise_fma(ai,  u,
                          __builtin_elementwise_fma(di, p2, p3));
        }
    }

    // Quaternions (scalar per body).
    float qw0, qx0, qy0, qz0, qw1, qx1, qy1, qz1;
    quat_from_R(R[0][0].x, R[0][1].x, R[0][2].x,
                R[1][0].x, R[1][1].x, R[1][2].x,
                R[2][0].x, R[2][1].x, R[2][2].x, qw0, qx0, qy0, qz0);
    quat_from_R(R[0][0].y, R[0][1].y, R[0][2].y,
                R[1][0].y, R[1][1].y, R[1][2].y,
                R[2][0].y, R[2][1].y, R[2][2].y, qw1, qx1, qy1, qz1);

    // ---- stage outputs: 14 contiguous floats per thread ----
    {
        float* o = lds + tid * 14;
        o[0]  = R[0][3].x; o[1]  = R[1][3].x; o[2]  = R[2][3].x;
        o[3]  = qw0; o[4]  = qx0; o[5]  = qy0; o[6]  = qz0;
        o[7]  = R[0][3].y; o[8]  = R[1][3].y; o[9]  = R[2][3].y;
        o[10] = qw1; o[11] = qx1; o[12] = qy1; o[13] = qz1;
    }

    __syncthreads();

    const long long outBase = (long long)tileBase * 7;
    float* __restrict__ gout = out + outBase;

    if (fullTile) {
        // 3584 floats = 896 float4; last iteration is wave-uniform (128 lanes = 4 waves).
#if USE_ASYNC
#pragma unroll
        for (int i = 0; i < 4; ++i) {
            const int f4 = i * TPB + tid;
            if (f4 < 896) {
                __builtin_amdgcn_global_store_async_from_lds_b128(
                    (AS1 v4i*)(gout + 4 * f4), (AS3 v4i*)(lds + 4 * f4), 0, 0);
            }
        }
        __builtin_amdgcn_s_wait_asynccnt(0);
#else
#pragma unroll
        for (int i = 0; i < 4; ++i) {
            const int f4 = i * TPB + tid;
            if (f4 < 896) ((float4*)gout)[f4] = ((const float4*)lds)[f4];
        }
#endif
    } else {
        const int nOut = (B - tileBase) * 7;
#pragma unroll
        for (int i = 0; i < 14; ++i) {
            const int f = i * TPB + tid;
            if (f < nOut) gout[f] = lds[f];
        }
    }
}

extern "C" void kernel_launch(void* const* d_in, const int* in_sizes, int n_in,
                              void* d_out, int out_size, void* d_ws, size_t ws_size,
                              hipStream_t stream) {
    (void)n_in; (void)out_size; (void)d_ws; (void)ws_size;
    const float* ja = (const float*)d_in[0];
    const float* a  = (const float*)d_in[1];
    const float* d  = (const float*)d_in[2];
    const float* al = (const float*)d_in[3];
    float* out = (float*)d_out;

    const int B = in_sizes[0] / 6;
    const int blocks = (B + TILE - 1) / TILE;
    fk_kernel<<<dim3(blocks), dim3(TPB), 0, stream>>>(ja, a, d, al, out, B);
}